// SlicedWassersteinLoss_69080253989043
// MI455X (gfx1250) — compile-verified
//
#include <hip/hip_runtime.h>
#include <hip/hip_bf16.h>
#include <math.h>

// Problem constants from the reference
#define NPTS 65536
#define DDIM 512
#define PPROJ 100
#define PPAD 112          // 7 tiles of 16 (100 padded)
#define PTILES 7

typedef __attribute__((ext_vector_type(2))) float v2f;
typedef __attribute__((ext_vector_type(8))) float v8f;

// ---------------------------------------------------------------------------
// Kernel 1: inverse norms of projection columns (tiny)
// ---------------------------------------------------------------------------
__global__ __launch_bounds__(128)
void swl_norm_kernel(const float* __restrict__ proj, float* __restrict__ inv_norm) {
    int p = blockIdx.x * blockDim.x + threadIdx.x;
    if (p >= PPROJ) return;
    float s = 0.0f;
    for (int d = 0; d < DDIM; ++d) {
        float v = proj[(size_t)d * PPROJ + p];
        s += v * v;
    }
    float n = sqrtf(s);
    inv_norm[p] = 1.0f / fmaxf(n, 1e-12f);
}

// ---------------------------------------------------------------------------
// Kernel 2: GEMM with V_WMMA_F32_16X16X4_F32 and full A-reuse.
//   outT[p][m] = sum_d A[m][d] * proj[d][p] * inv_norm[p]
// The ENTIRE normalized projection slice (512 x 112 fp32 = 224 KB) is staged
// in the CDNA5 320 KB WGP LDS, so each A fragment is loaded from HBM exactly
// once and feeds 7 back-to-back WMMAs (one per 16-column tile).
// blockIdx.y selects x vs y so each block streams only one source matrix
// (HBM traffic stays at the 268 MB minimum; accumulators stay at 7x8 VGPRs).
// Fragment layouts per CDNA5 ISA 7.12.2:
//   A 16x4 : lanes 0-15 hold {A[m][k+0],A[m][k+1]}, lanes 16-31 {k+2,k+3}
//   B 4x16 : lanes 0-15 hold {B[k+0][n],B[k+1][n]}, lanes 16-31 {k+2,k+3}
//   C 16x16: VGPR v -> M = v (lanes 0-15) / v+8 (lanes 16-31), N = lane&15
// ---------------------------------------------------------------------------
__global__ __launch_bounds__(256)
void swl_gemm_kernel(const float* __restrict__ x, const float* __restrict__ y,
                     const float* __restrict__ proj, const float* __restrict__ inv_norm,
                     float* __restrict__ xT, float* __restrict__ yT) {
    extern __shared__ float Bs[];   // [DDIM][PPAD] = 512*112 fp32 = 224 KB

    // Stage full normalized B (zero-pad columns >= PPROJ). proj is ~200 KB
    // total so these re-reads are L2-resident (192 MB L2).
    for (int idx = threadIdx.x; idx < DDIM * PPAD; idx += 256) {
        int d = idx / PPAD;
        int p = idx - d * PPAD;
        float v = 0.0f;
        if (p < PPROJ) v = proj[(size_t)d * PPROJ + p] * inv_norm[p];
        Bs[idx] = v;
    }
    __syncthreads();

    const int wave = threadIdx.x >> 5;      // 0..7 (wave32)
    const int lane = threadIdx.x & 31;
    const int half = lane >> 4;             // 0 or 1
    const int l15  = lane & 15;
    const int m0   = blockIdx.x * 128 + wave * 16;

    const bool   isY  = (blockIdx.y != 0);
    const float* src  = isY ? y : x;
    float*       dstT = isY ? yT : xT;

    const float* arow = src + (size_t)(m0 + l15) * DDIM;

    v8f acc[PTILES];
    #pragma unroll
    for (int t = 0; t < PTILES; ++t) acc[t] = (v8f){};

    for (int k = 0; k < DDIM; k += 4) {
        const int ka = k + half * 2;
        v2f a = *(const v2f*)(arow + ka);            // one HBM fragment ...
        #pragma unroll
        for (int t = 0; t < PTILES; ++t) {           // ... feeds 7 WMMAs
            const int pc = t * 16 + l15;
            v2f b;
            b.x = Bs[(ka + 0) * PPAD + pc];
            b.y = Bs[(ka + 1) * PPAD + pc];
            acc[t] = __builtin_amdgcn_wmma_f32_16x16x4_f32(
                false, a, false, b, (short)0, acc[t], false, false);
        }
    }

    // Store transposed: column p contiguous over N for the sort phase.
    #pragma unroll
    for (int t = 0; t < PTILES; ++t) {
        const int n = t * 16 + l15;
        if (n < PPROJ) {
            float* col = dstT + (size_t)n * NPTS;
            #pragma unroll
            for (int v = 0; v < 8; ++v) {
                int m = m0 + v + half * 8;
                col[m] = acc[t][v];
            }
        }
    }
}

// ---------------------------------------------------------------------------
// Kernel 3: full-column bitonic sort entirely in LDS.
// One 65536-float column = 256 KB -> fits in the CDNA5 320 KB WGP LDS.
// 200 blocks (100 x-columns then 100 y-columns), 1024 threads each.
// ---------------------------------------------------------------------------
__global__ __launch_bounds__(1024)
void swl_bitonic_kernel(float* __restrict__ xT, float* __restrict__ yT) {
    extern __shared__ float smem[];   // 65536 floats = 256 KB dynamic LDS

    const int b = blockIdx.x;
    float* base = (b < PPROJ) ? (xT + (size_t)b * NPTS)
                              : (yT + (size_t)(b - PPROJ) * NPTS);

    for (int i = threadIdx.x; i < NPTS; i += 1024) smem[i] = base[i];
    __syncthreads();

    for (int k = 2; k <= NPTS; k <<= 1) {
        for (int j = k >> 1; j > 0; j >>= 1) {
            for (int t = threadIdx.x; t < NPTS; t += 1024) {
                int ixj = t ^ j;
                if (ixj > t) {
                    float a = smem[t];
                    float c = smem[ixj];
                    bool up = ((t & k) == 0);
                    if ((a > c) == up) { smem[t] = c; smem[ixj] = a; }
                }
            }
            __syncthreads();
        }
    }

    for (int i = threadIdx.x; i < NPTS; i += 1024) base[i] = smem[i];
}

// ---------------------------------------------------------------------------
// Kernel 4: per-projection sum of squared diffs of sorted columns.
// Fixed-order strided accumulation + LDS tree -> deterministic.
// ---------------------------------------------------------------------------
__global__ __launch_bounds__(256)
void swl_wd_kernel(const float* __restrict__ xT, const float* __restrict__ yT,
                   float* __restrict__ wd) {
    __shared__ float red[256];
    const int p = blockIdx.x;
    const float* xs = xT + (size_t)p * NPTS;
    const float* ys = yT + (size_t)p * NPTS;

    float acc = 0.0f;
    for (int i = threadIdx.x; i < NPTS; i += 256) {
        float d = xs[i] - ys[i];
        acc += d * d;
    }
    red[threadIdx.x] = acc;
    __syncthreads();
    for (int s = 128; s > 0; s >>= 1) {
        if ((int)threadIdx.x < s) red[threadIdx.x] += red[threadIdx.x + s];
        __syncthreads();
    }
    if (threadIdx.x == 0) wd[p] = red[0];   // = NPTS * wd_mean[p]
}

// ---------------------------------------------------------------------------
// Kernel 5: final scalar = sqrt( sum_p wd_sum[p] / (N*P) )
// ---------------------------------------------------------------------------
__global__ __launch_bounds__(128)
void swl_final_kernel(const float* __restrict__ wd, float* __restrict__ out) {
    __shared__ float red[128];
    float v = ((int)threadIdx.x < PPROJ) ? wd[threadIdx.x] : 0.0f;
    red[threadIdx.x] = v;
    __syncthreads();
    for (int s = 64; s > 0; s >>= 1) {
        if ((int)threadIdx.x < s) red[threadIdx.x] += red[threadIdx.x + s];
        __syncthreads();
    }
    if (threadIdx.x == 0)
        out[0] = sqrtf(red[0] / ((float)NPTS * (float)PPROJ));
}

// ---------------------------------------------------------------------------
// Host-side launcher (graph-capture safe: only kernel launches on `stream`).
// ---------------------------------------------------------------------------
extern "C" void kernel_launch(void* const* d_in, const int* in_sizes, int n_in,
                              void* d_out, int out_size, void* d_ws, size_t ws_size,
                              hipStream_t stream) {
    const float* x    = (const float*)d_in[0];   // (N, D) fp32
    const float* y    = (const float*)d_in[1];   // (N, D) fp32
    const float* proj = (const float*)d_in[2];   // (D, P) fp32
    float* out = (float*)d_out;                  // scalar fp32

    // Workspace layout
    char* ws = (char*)d_ws;
    float* inv_norm = (float*)ws;                               // 256 floats (padded)
    float* xT = (float*)(ws + 1024);                            // [P][N] fp32
    float* yT = xT + (size_t)PPROJ * NPTS;                      // [P][N] fp32
    float* wd = yT + (size_t)PPROJ * NPTS;                      // [P] fp32
    // Total: 1024 + 2*100*65536*4 + 400 bytes  (~52.4 MB)

    swl_norm_kernel<<<1, 128, 0, stream>>>(proj, inv_norm);

    // One block row-range per 128 rows; blockIdx.y = {x, y}. Each block loads
    // its source rows from HBM exactly once (full projection reuse via LDS).
    dim3 ggrid(NPTS / 128, 2);                                  // (512, 2)
    swl_gemm_kernel<<<ggrid, 256, DDIM * PPAD * sizeof(float), stream>>>(
        x, y, proj, inv_norm, xT, yT);

    swl_bitonic_kernel<<<2 * PPROJ, 1024, NPTS * sizeof(float), stream>>>(xT, yT);

    swl_wd_kernel<<<PPROJ, 256, 0, stream>>>(xT, yT, wd);

    swl_final_kernel<<<1, 128, 0, stream>>>(wd, out);
}